// gcSoftmax_54065048322743
// MI455X (gfx1250) — compile-verified
//
#include <hip/hip_runtime.h>
#include <stdint.h>

#define N_NODES 100000
#define BLOCK   256
#define EPL     4                 // edges per lane per tile
#define TILE    (BLOCK * EPL)     // 1024 edges per tile

// ---------------------------------------------------------------------------
// CDNA5 (gfx1250) inline-asm helpers.
//
// - global_load_async_to_lds_b128: VFLAT async op, ASYNCcnt-tracked
//   (CDNA5 ISA). VDST = VGPR holding the LDS byte
//   address (low 32 bits of a generic shared pointer ARE the LDS offset).
// - global_atomic_add_f32 (no-return form, scope:SCOPE_DEV): guarantees the
//   native fire-and-forget f32 RMW at device scope — avoids any CAS-loop
//   fallback, which would be disastrous at ~64 colliding edges per node.
// ---------------------------------------------------------------------------
__device__ __forceinline__ void async_load_b128(uint32_t lds_byte_addr, const void* gaddr) {
    asm volatile("global_load_async_to_lds_b128 %0, %1, off"
                 :: "v"(lds_byte_addr), "v"(gaddr) : "memory");
}
__device__ __forceinline__ void s_wait_asynccnt_0() {
    asm volatile("s_wait_asynccnt 0x0" ::: "memory");
}
__device__ __forceinline__ void s_wait_asynccnt_3() {
    asm volatile("s_wait_asynccnt 0x3" ::: "memory");
}
__device__ __forceinline__ void atomic_add_f32_dev(float* p, float v) {
    asm volatile("global_atomic_add_f32 %0, %1, off scope:SCOPE_DEV"
                 :: "v"(p), "v"(v) : "memory");
}

// ---------------------------------------------------------------------------
// Pass 0: zero the denom accumulator (400 KB, L2-resident thereafter).
// ---------------------------------------------------------------------------
__global__ void __launch_bounds__(BLOCK)
gc_zero(float* __restrict__ denom, int n) {
    int i = blockIdx.x * BLOCK + threadIdx.x;
    if (i < n) denom[i] = 0.0f;
}

// ---------------------------------------------------------------------------
// Pass 1 (fused): pass-through copy of edge_index into d_out AND
// denom[dst[e]] += exp(w[e]).  Reading both index rows exactly once here
// saves a second stream of the dst row vs. a separate D2D memcpy
// (~51 MB less traffic). Atomics are non-returning global_atomic_add_f32
// into the L2-resident denom array.
// ---------------------------------------------------------------------------
__global__ void __launch_bounds__(BLOCK)
gc_accum_copy(const long long* __restrict__ src, const long long* __restrict__ dst,
              const float* __restrict__ w, float* __restrict__ denom,
              long long* __restrict__ out_idx, int E) {
    const long stride = (long)gridDim.x * BLOCK * 4;
    const long E4     = (long)E & ~3L;
    const long EL     = (long)E;

    for (long i = ((long)blockIdx.x * BLOCK + threadIdx.x) * 4; i < E4; i += stride) {
        longlong4 sv = *(const longlong4*)(src + i);
        longlong4 dv = *(const longlong4*)(dst + i);
        float4    wv = *(const float4*)(w + i);

        *(longlong4*)(out_idx + i)      = sv;   // edge_index[0] pass-through
        *(longlong4*)(out_idx + EL + i) = dv;   // edge_index[1] pass-through

        atomic_add_f32_dev(&denom[(int)dv.x], __expf(wv.x));
        atomic_add_f32_dev(&denom[(int)dv.y], __expf(wv.y));
        atomic_add_f32_dev(&denom[(int)dv.z], __expf(wv.z));
        atomic_add_f32_dev(&denom[(int)dv.w], __expf(wv.w));
    }
    for (long i = E4 + blockIdx.x * BLOCK + threadIdx.x; i < EL;
         i += (long)gridDim.x * BLOCK) {
        out_idx[i]      = src[i];
        out_idx[EL + i] = dst[i];
        atomic_add_f32_dev(&denom[(int)dst[i]], __expf(w[i]));
    }
}

// ---------------------------------------------------------------------------
// Pass 2: out[e] = exp(w[e]) * rcp(denom[dst[e]]).
// The w/dst streams are staged through LDS with double-buffered async
// global->LDS b128 copies (3 async ops per lane per tile, ASYNCcnt-tracked),
// overlapping the sequential HBM stream with the random L2 denom gathers.
// ---------------------------------------------------------------------------
__global__ void __launch_bounds__(BLOCK)
gc_normalize(const long long* __restrict__ dst, const float* __restrict__ w,
             const float* __restrict__ denom, float* __restrict__ out, int E) {
    __shared__ float     s_w[2][TILE];
    __shared__ long long s_d[2][TILE];

    const int  t      = threadIdx.x;
    const long ntiles = (long)E / TILE;          // full tiles only
    const long stride = gridDim.x;
    long       tile   = blockIdx.x;

    const uint32_t lw0 = (uint32_t)(size_t)&s_w[0][t * EPL];
    const uint32_t lw1 = (uint32_t)(size_t)&s_w[1][t * EPL];
    const uint32_t ld0 = (uint32_t)(size_t)&s_d[0][t * EPL];
    const uint32_t ld1 = (uint32_t)(size_t)&s_d[1][t * EPL];

    auto issue = [&](int buf, long tl) {
        const long base = tl * TILE + (long)t * EPL;
        const uint32_t lw = buf ? lw1 : lw0;
        const uint32_t ld = buf ? ld1 : ld0;
        async_load_b128(lw,      (const void*)(w   + base));      // 4 floats
        async_load_b128(ld,      (const void*)(dst + base));      // 2 int64
        async_load_b128(ld + 16, (const void*)(dst + base + 2));  // 2 int64
    };

    int buf = 0;
    if (tile < ntiles) issue(0, tile);

    for (; tile < ntiles; tile += stride) {
        const long next = tile + stride;
        if (next < ntiles) {
            issue(buf ^ 1, next);
            s_wait_asynccnt_3();   // 3 newest outstanding belong to `next`
        } else {
            s_wait_asynccnt_0();
        }

        const long base = tile * TILE + (long)t * EPL;
        float e0 = __expf(s_w[buf][t * EPL + 0]);
        float e1 = __expf(s_w[buf][t * EPL + 1]);
        float e2 = __expf(s_w[buf][t * EPL + 2]);
        float e3 = __expf(s_w[buf][t * EPL + 3]);
        int d0 = (int)s_d[buf][t * EPL + 0];
        int d1 = (int)s_d[buf][t * EPL + 1];
        int d2 = (int)s_d[buf][t * EPL + 2];
        int d3 = (int)s_d[buf][t * EPL + 3];

        float4 r;
        r.x = e0 * __builtin_amdgcn_rcpf(denom[d0]);
        r.y = e1 * __builtin_amdgcn_rcpf(denom[d1]);
        r.z = e2 * __builtin_amdgcn_rcpf(denom[d2]);
        r.w = e3 * __builtin_amdgcn_rcpf(denom[d3]);
        *(float4*)(out + base) = r;
        buf ^= 1;
    }

    // tail (E % TILE) via direct loads
    for (long i = ntiles * TILE + (long)blockIdx.x * BLOCK + t; i < E;
         i += (long)gridDim.x * BLOCK) {
        out[i] = __expf(w[i]) * __builtin_amdgcn_rcpf(denom[(int)dst[i]]);
    }
}

// ---------------------------------------------------------------------------
// Launch: zero -> fused copy+accumulate -> normalize, serialized on `stream`.
// ---------------------------------------------------------------------------
extern "C" void kernel_launch(void* const* d_in, const int* in_sizes, int n_in,
                              void* d_out, int out_size, void* d_ws, size_t ws_size,
                              hipStream_t stream) {
    const long long* edge_index = (const long long*)d_in[0];   // [2][E] int64
    const float*     w          = (const float*)d_in[1];       // [E] fp32
    const int        E          = in_sizes[1];
    const long long* srcrow     = edge_index;                  // edge_index[0]
    const long long* dstrow     = edge_index + E;              // edge_index[1]
    float*           denom      = (float*)d_ws;                // 100K fp32 in L2

    gc_zero<<<(N_NODES + BLOCK - 1) / BLOCK, BLOCK, 0, stream>>>(denom, N_NODES);

    long long* out_idx = (long long*)d_out;
    gc_accum_copy<<<3072, BLOCK, 0, stream>>>(srcrow, dstrow, w, denom, out_idx, E);

    float* outw = (float*)((char*)d_out + (size_t)2 * (size_t)E * sizeof(long long));
    long ntiles = (long)E / TILE;
    int  nblk   = (int)(ntiles < 1 ? 1 : (ntiles > 3072 ? 3072 : ntiles));
    gc_normalize<<<nblk, BLOCK, 0, stream>>>(dstrow, w, denom, outw, E);
}